// RWNNLM_48550310314126
// MI455X (gfx1250) — compile-verified
//
#include <hip/hip_runtime.h>
#include <hip/hip_bf16.h>

typedef __attribute__((ext_vector_type(2))) float v2f;
typedef __attribute__((ext_vector_type(8))) float v8f;

#define LDV 3488          // activation slab leading dim (nodes excl. final vocab outputs)
#define BATCH_ 512
#define EDGE_PROB 0.03
#define MBLK 4            // 16x16 C tiles per wave, stacked along M (64x16 per wave)

// ---------------------------------------------------------------------------
// Kernel 1: deterministic graph rebuild (numpy RandomState(0) / MT19937) +
// densification of the sparse edge weights into zero-padded per-layer dense W.
// Single workgroup; twist parallelized into 3 barrier stages (recurrence
// distance 227), tempering + double-compare fully lane-parallel.
// ---------------------------------------------------------------------------
__global__ __launch_bounds__(1024) void RWNN_build_kernel(
    const float* __restrict__ weights,
    float* __restrict__ Wd,            // dense weights, all layers concatenated
    unsigned char* __restrict__ mask)  // scratch mask, >= 3488*512 bytes
{
  constexpr int START_[6] = {2050, 2338, 2626, 2914, 3201, 3488};
  constexpr int SIZE_[6]  = {288, 288, 288, 287, 287, 512};
  constexpr int NPAD_[6]  = {288, 288, 288, 288, 288, 512};
  constexpr int KPAD_[6]  = {2052, 2340, 2628, 2916, 3204, 3488};
  constexpr int WOFF_[6]  = {0, 590976, 1264896, 2021760, 2861568, 3784320};

  __shared__ unsigned mt[624], mo[624], tw[624];
  __shared__ int s_bpos, s_dcount, s_wbase, s_E, s_nempty;
  __shared__ int s_rowoff[3488];
  const int tid = threadIdx.x;

  // numpy legacy scalar seeding: init_genrand(0)
  if (tid == 0) {
    mt[0] = 0u;
    for (int i = 1; i < 624; ++i)
      mt[i] = 1812433253u * (mt[i - 1] ^ (mt[i - 1] >> 30)) + (unsigned)i;
    s_bpos = 624;   // force twist before first output
    s_wbase = 0;
  }
  __syncthreads();

  auto twist_parallel = [&]() {
    for (int i = tid; i < 624; i += 1024) mo[i] = mt[i];
    __syncthreads();
    if (tid < 227) {  // i in [0,227): all-old inputs
      unsigned y = (mo[tid] & 0x80000000u) | (mo[tid + 1] & 0x7fffffffu);
      mt[tid] = mo[tid + 397] ^ (y >> 1) ^ ((y & 1u) ? 0x9908b0dfu : 0u);
    }
    __syncthreads();
    if (tid < 227) {  // i in [227,454): needs new[i-227] from stage 1
      int i = tid + 227;
      unsigned y = (mo[i] & 0x80000000u) | (mo[i + 1] & 0x7fffffffu);
      mt[i] = mt[i - 227] ^ (y >> 1) ^ ((y & 1u) ? 0x9908b0dfu : 0u);
    }
    __syncthreads();
    if (tid < 170) {  // i in [454,624): needs new[i-227] from stage 2 (and new mt[0] for i=623)
      int i = tid + 454;
      unsigned nxt = (i == 623) ? mt[0] : mo[i + 1];
      unsigned y = (mo[i] & 0x80000000u) | (nxt & 0x7fffffffu);
      mt[i] = mt[i - 227] ^ (y >> 1) ^ ((y & 1u) ? 0x9908b0dfu : 0u);
    }
    __syncthreads();
    if (tid < 624) {  // temper
      unsigned y = mt[tid];
      y ^= y >> 11;
      y ^= (y << 7) & 0x9d2c5680u;
      y ^= (y << 15) & 0xefc60000u;
      y ^= y >> 18;
      tw[tid] = y;
    }
    __syncthreads();
  };

  // sequential (thread-0 only) raw draw for the never-in-practice randint path
  auto seq_next = [&]() -> unsigned {
    if (s_bpos >= 624) {
      for (int k = 0; k < 227; ++k) {
        unsigned y = (mt[k] & 0x80000000u) | (mt[k + 1] & 0x7fffffffu);
        mt[k] = mt[k + 397] ^ (y >> 1) ^ ((y & 1u) ? 0x9908b0dfu : 0u);
      }
      for (int k = 227; k < 623; ++k) {
        unsigned y = (mt[k] & 0x80000000u) | (mt[k + 1] & 0x7fffffffu);
        mt[k] = mt[k - 227] ^ (y >> 1) ^ ((y & 1u) ? 0x9908b0dfu : 0u);
      }
      {
        unsigned y = (mt[623] & 0x80000000u) | (mt[0] & 0x7fffffffu);
        mt[623] = mt[396] ^ (y >> 1) ^ ((y & 1u) ? 0x9908b0dfu : 0u);
      }
      for (int i = 0; i < 624; ++i) {
        unsigned y = mt[i];
        y ^= y >> 11; y ^= (y << 7) & 0x9d2c5680u;
        y ^= (y << 15) & 0xefc60000u; y ^= y >> 18;
        tw[i] = y;
      }
      s_bpos = 0;
    }
    return tw[s_bpos++];
  };

  for (int l = 0; l < 6; ++l) {
    const int START = START_[l], SIZE = SIZE_[l];
    const int NPAD = NPAD_[l], KPAD = KPAD_[l];
    const int ND = START * SIZE;  // doubles needed this layer
    if (tid == 0) s_dcount = 0;
    __syncthreads();

    // ---- phase 1: mask[r,c] = (uniform() < EDGE_P), row-major ----
    for (;;) {
      if (s_dcount >= ND) break;
      if (s_bpos >= 624) {
        twist_parallel();
        if (tid == 0) s_bpos = 0;
        __syncthreads();
      }
      int bpos = s_bpos, dc = s_dcount;
      int npairs = (624 - bpos) >> 1;            // block length even, pairs never straddle
      int rem = ND - dc;
      if (npairs > rem) npairs = rem;
      if (tid < npairs) {
        unsigned a = tw[bpos + 2 * tid] >> 5;    // 27 bits
        unsigned b = tw[bpos + 2 * tid + 1] >> 6;// 26 bits
        double d = ((double)a * 67108864.0 + (double)b) *
                   (1.0 / 9007199254740992.0);
        mask[dc + tid] = (d < EDGE_PROB) ? (unsigned char)1 : (unsigned char)0;
      }
      __syncthreads();
      if (tid == 0) { s_bpos = bpos + 2 * npairs; s_dcount = dc + npairs; }
      __syncthreads();
    }
    __syncthreads();

    // ---- phase 2: empty-column fix (P(empty) ~ e^-62; kept for fidelity) ----
    if (tid == 0) s_nempty = 0;
    __syncthreads();
    for (int c = tid; c < SIZE; c += 1024) {
      int cnt = 0;
      for (int r = 0; r < START; ++r) cnt += mask[r * SIZE + c];
      if (cnt == 0) atomicAdd(&s_nempty, 1);
    }
    __syncthreads();
    if (s_nempty > 0) {
      if (tid == 0) {
        unsigned rngmax = (unsigned)(START - 1);
        unsigned m = rngmax;
        m |= m >> 1; m |= m >> 2; m |= m >> 4; m |= m >> 8; m |= m >> 16;
        for (int c = 0; c < SIZE; ++c) {
          int cnt = 0;
          for (int r = 0; r < START; ++r) cnt += mask[r * SIZE + c];
          if (cnt == 0) {
            unsigned long long v;
            do {
              unsigned hi = seq_next();
              unsigned lo = seq_next();
              v = ((((unsigned long long)hi << 32) | lo)) & (unsigned long long)m;
            } while (v > (unsigned long long)rngmax);
            mask[(int)v * SIZE + c] = 1;
          }
        }
      }
      __syncthreads();
    }

    // ---- phase 3: row counts -> exclusive prefix -> densify ----
    for (int r = tid; r < START; r += 1024) {
      int cnt = 0;
      const unsigned char* mrow = mask + (size_t)r * SIZE;
      for (int c = 0; c < SIZE; ++c) cnt += mrow[c];
      s_rowoff[r] = cnt;
    }
    __syncthreads();
    if (tid == 0) {
      int run = 0;
      for (int r = 0; r < START; ++r) { int c = s_rowoff[r]; s_rowoff[r] = run; run += c; }
      s_E = run;
    }
    __syncthreads();
    float* Wl = Wd + (size_t)WOFF_[l];
    for (int r = tid; r < KPAD; r += 1024) {
      if (r < START) {
        int e = s_wbase + s_rowoff[r];
        const unsigned char* mrow = mask + (size_t)r * SIZE;
        for (int c = 0; c < NPAD; ++c) {
          float w = 0.0f;
          if (c < SIZE && mrow[c]) w = weights[e++];
          Wl[(size_t)r * NPAD + c] = w;
        }
      } else {  // zero K-padding rows
        for (int c = 0; c < NPAD; ++c) Wl[(size_t)r * NPAD + c] = 0.0f;
      }
    }
    __syncthreads();
    if (tid == 0) s_wbase += s_E;
    __syncthreads();
  }
}

// ---------------------------------------------------------------------------
// Kernel 2: embedding lookup + positional add + bias columns into the slab.
// ---------------------------------------------------------------------------
__global__ void RWNN_embed_kernel(const int* __restrict__ ids,
                                  const float* __restrict__ tok,
                                  const float* __restrict__ pos,
                                  float* __restrict__ vals)
{
  int idx = blockIdx.x * blockDim.x + threadIdx.x;
  const int TOT = BATCH_ * 2050;
  if (idx >= TOT) return;
  int b = idx / 2050;
  int c = idx - b * 2050;
  float v;
  if (c < 2048) {
    int t = c >> 5, d = c & 31;
    v = tok[(size_t)ids[b * 64 + t] * 32 + d] + pos[c];
  } else {
    v = 1.0f;  // bias node
  }
  vals[(size_t)b * LDV + c] = v;
}

// ---------------------------------------------------------------------------
// Kernel 3: per-layer dense GEMM with V_WMMA_F32_16X16X4_F32.
// One wave computes MBLK=4 16x16 C tiles stacked along M (64x16 of output),
// reusing each (scattered, non-vectorizable) B fetch across 4 WMMAs:
// loads/WMMA drops from 3 to 1.5.
// A 16x4 layout: v0={K0|K2}, v1={K1|K3}; lanes 0-15 / 16-31 both hold M=0..15.
// B 4x16 layout: v0={row K0|row K2}, v1={row K1|row K3}; lanes hold N.
// C 16x16: vgpr j -> M=j (lanes 0-15) / M=j+8 (lanes 16-31), N=lane&15.
// ---------------------------------------------------------------------------
__global__ __launch_bounds__(256) void RWNN_gemm_kernel(
    const float* __restrict__ V, const float* __restrict__ W,
    float* __restrict__ out, int ldOut,
    int Kpad, int Npad, int Nvalid, int applyTanh)
{
  const int wid  = (blockIdx.x * blockDim.x + threadIdx.x) >> 5;
  const int lane = threadIdx.x & 31;
  const int tilesN = Npad >> 4;
  const int tileN = wid % tilesN;
  const int mgrp  = wid / tilesN;
  if (mgrp >= BATCH_ / (16 * MBLK)) return;   // uniform across the wave

  const int half = lane >> 4;   // 0: K+{0,1}, 1: K+{2,3}
  const int r16  = lane & 15;
  const int koff = half * 2;

  const float* aptr = V + (size_t)(mgrp * 16 * MBLK + r16) * LDV + koff;
  const float* bptr = W + (size_t)koff * Npad + (tileN * 16 + r16);

  v8f c0 = {}, c1 = {}, c2 = {}, c3 = {};
  for (int k = 0; k < Kpad; k += 4) {
    v2f b;
    b.x = bptr[(size_t)k * Npad];               // B[k+koff][n]
    b.y = bptr[(size_t)(k + 1) * Npad];         // B[k+koff+1][n]
    v2f a0 = *(const v2f*)(aptr + k);
    v2f a1 = *(const v2f*)(aptr + (size_t)16 * LDV + k);
    v2f a2 = *(const v2f*)(aptr + (size_t)32 * LDV + k);
    v2f a3 = *(const v2f*)(aptr + (size_t)48 * LDV + k);
    c0 = __builtin_amdgcn_wmma_f32_16x16x4_f32(false, a0, false, b, (short)0, c0, false, false);
    c1 = __builtin_amdgcn_wmma_f32_16x16x4_f32(false, a1, false, b, (short)0, c1, false, false);
    c2 = __builtin_amdgcn_wmma_f32_16x16x4_f32(false, a2, false, b, (short)0, c2, false, false);
    c3 = __builtin_amdgcn_wmma_f32_16x16x4_f32(false, a3, false, b, (short)0, c3, false, false);
  }

  const int n = tileN * 16 + r16;
  if (n < Nvalid) {
    v8f acc[MBLK] = {c0, c1, c2, c3};
#pragma unroll
    for (int t = 0; t < MBLK; ++t) {
      const int mbase = mgrp * 16 * MBLK + t * 16 + half * 8;
#pragma unroll
      for (int j = 0; j < 8; ++j) {
        float v = acc[t][j];
        if (applyTanh) v = tanhf(v);
        out[(size_t)(mbase + j) * ldOut + n] = v;
      }
    }
  }
}

// ---------------------------------------------------------------------------
extern "C" void kernel_launch(void* const* d_in, const int* in_sizes, int n_in,
                              void* d_out, int out_size, void* d_ws, size_t ws_size,
                              hipStream_t stream) {
  const int*   ids = (const int*)d_in[0];
  const float* tok = (const float*)d_in[1];
  const float* pos = (const float*)d_in[2];
  const float* wts = (const float*)d_in[3];
  float* out = (float*)d_out;

  // workspace layout (256B aligned segments)
  char* ws = (char*)d_ws;
  const size_t valsBytes = (size_t)BATCH_ * LDV * 4;     // 7,143,424
  const size_t wdBytes   = (size_t)5570176 * 4;          // 22,280,704
  float* vals = (float*)ws;
  float* Wd   = (float*)(ws + ((valsBytes + 255) & ~(size_t)255));
  unsigned char* mask =
      (unsigned char*)(ws + ((valsBytes + 255) & ~(size_t)255) +
                            ((wdBytes + 255) & ~(size_t)255));

  // 1) rebuild graph + densify weights (deterministic every call)
  RWNN_build_kernel<<<1, 1024, 0, stream>>>(wts, Wd, mask);

  // 2) embeddings + bias
  {
    int tot = BATCH_ * 2050;
    RWNN_embed_kernel<<<(tot + 255) / 256, 256, 0, stream>>>(ids, tok, pos, vals);
  }

  // 3) six WMMA GEMM layers
  const int START_[6] = {2050, 2338, 2626, 2914, 3201, 3488};
  const int SIZE_[6]  = {288, 288, 288, 287, 287, 512};
  const int NPAD_[6]  = {288, 288, 288, 288, 288, 512};
  const int KPAD_[6]  = {2052, 2340, 2628, 2916, 3204, 3488};
  const int WOFF_[6]  = {0, 590976, 1264896, 2021760, 2861568, 3784320};

  for (int l = 0; l < 6; ++l) {
    const int tiles = (NPAD_[l] / 16) * (BATCH_ / (16 * MBLK));
    const int blocks = (tiles + 7) / 8;  // 8 waves per 256-thread block
    float* dst  = (l < 5) ? (vals + START_[l]) : out;
    int    ldO  = (l < 5) ? LDV : 512;
    int    doTanh = (l < 5) ? 1 : 0;
    RWNN_gemm_kernel<<<blocks, 256, 0, stream>>>(
        vals, Wd + (size_t)WOFF_[l], dst, ldO,
        KPAD_[l], NPAD_[l], SIZE_[l], doTanh);
  }
}